// StandardROIHeads_2267742732669
// MI455X (gfx1250) — compile-verified
//
#include <hip/hip_runtime.h>
#include <hip/hip_bf16.h>

// Problem constants (match reference)
#define NB 16
#define NP 4096
#define NG 512
#define NS 512
#define NUM_FG 128
#define NUM_BG 384
#define NUM_CLASSES 80

#define P1_THREADS 256
#define P2_THREADS 512

// ---------------------------------------------------------------------------
// Phase 1: per-proposal matched IoU max + argmax over all gt boxes.
// gt boxes are staged into LDS via CDNA5 async-to-LDS DMA (ASYNCcnt path).
// ---------------------------------------------------------------------------
__global__ void __launch_bounds__(P1_THREADS)
roi_match_kernel(const float4* __restrict__ prop,    // [NB*NP] xyxy
                 const float4* __restrict__ gt,      // [NB*NG] xyxy
                 float* __restrict__ matched_vals,   // [NB*NP]
                 int* __restrict__ matched_idxs)     // [NB*NP]
{
    const int b = blockIdx.y;
    const int p = blockIdx.x * P1_THREADS + threadIdx.x;

    __shared__ float4 s_gt[NG];   // 8 KB
    __shared__ float  s_at[NG];   // 2 KB

    // ---- async DMA stage: 512 float4 (8 KB) of gt boxes into LDS ----
    const float4* src = gt + b * NG;
    for (int g = threadIdx.x; g < NG; g += P1_THREADS) {
        unsigned int lds_off =
            (unsigned int)(unsigned long long)(&s_gt[g]);     // addr[31:0] == LDS offset
        unsigned long long gaddr = (unsigned long long)(const void*)(src + g);
        asm volatile("global_load_async_to_lds_b128 %0, %1, off"
                     :: "v"(lds_off), "v"(gaddr) : "memory");
    }
    asm volatile("s_wait_asynccnt 0" ::: "memory");
    __syncthreads();

    // gt areas (reads are LDS broadcast-friendly later)
    for (int g = threadIdx.x; g < NG; g += P1_THREADS) {
        float4 t = s_gt[g];
        s_at[g] = (t.z - t.x) * (t.w - t.y);
    }
    __syncthreads();

    const float4 pr = prop[b * NP + p];
    const float  ap = (pr.z - pr.x) * (pr.w - pr.y);

    float best = -1.0f;   // all IoUs are >= 0, so g=0 always updates (argmax -> first max)
    int   bidx = 0;
#pragma unroll 4
    for (int g = 0; g < NG; ++g) {
        float4 t = s_gt[g];                       // same address across lanes -> broadcast
        float lx = fmaxf(t.x, pr.x);
        float ly = fmaxf(t.y, pr.y);
        float rx = fminf(t.z, pr.z);
        float ry = fminf(t.w, pr.w);
        float w  = fmaxf(rx - lx, 0.0f);
        float h  = fmaxf(ry - ly, 0.0f);
        float inter = w * h;
        float uni   = s_at[g] + ap - inter;
        float iou   = (inter > 0.0f) ? (inter / fmaxf(uni, 1e-8f)) : 0.0f;
        if (iou > best) { best = iou; bidx = g; }   // strict > == first-max (jnp.argmax)
    }
    matched_vals[b * NP + p] = best;
    matched_idxs[b * NP + p] = bidx;
}

// ---------------------------------------------------------------------------
// Phase 2: per-image subsample (single bitonic sort) + gather + output write.
// Key layout (ascending sort):
//   bit 62      : 0 = fg, 1 = bg            (fg entries first)
//   bits 46..16 : ~(prio_bits | 0x80000000) (priority descending; prio >= 0)
//   bits 15..0  : proposal index            (ties -> lowest index first, top_k rule)
// Slots whose reference score would be NEG_INF are masked in the output, so
// their gathered index is irrelevant -> one sort serves both fg and bg lists.
// ---------------------------------------------------------------------------
__global__ void __launch_bounds__(P2_THREADS)
roi_sample_kernel(const float4* __restrict__ prop,
                  const float4* __restrict__ gt,
                  const int*    __restrict__ gt_classes,   // [NB*NG]
                  const float*  __restrict__ prio,         // [NB*NP]
                  const float*  __restrict__ matched_vals, // [NB*NP]
                  const int*    __restrict__ matched_idxs, // [NB*NP]
                  float*        __restrict__ out)          // [NB*NS*8] then [NB*NS] cls
{
    const int b   = blockIdx.x;
    const int tid = threadIdx.x;

    __shared__ unsigned long long keys[NP];   // 32 KB
    __shared__ int sF;

    // build keys
    for (int i = tid; i < NP; i += P2_THREADS) {
        float sc = prio[b * NP + i];                       // in [0,1): sign bit 0
        bool  fg = matched_vals[b * NP + i] >= 0.5f;
        unsigned int u   = __float_as_uint(sc);
        unsigned int tpr = ~(u | 0x80000000u);             // 31 bits, larger prio -> smaller
        unsigned long long key =
            ((unsigned long long)(fg ? 0u : 1u) << 62) |
            ((unsigned long long)tpr << 16) |
            (unsigned long long)(unsigned int)i;
        keys[i] = key;
    }
    __syncthreads();

    // bitonic sort, ascending
    for (unsigned k = 2; k <= NP; k <<= 1) {
        for (unsigned j = k >> 1; j > 0; j >>= 1) {
            for (unsigned t = tid; t < NP; t += P2_THREADS) {
                unsigned ixj = t ^ j;
                if (ixj > t) {
                    bool up = ((t & k) == 0);
                    unsigned long long a = keys[t];
                    unsigned long long c = keys[ixj];
                    if ((a > c) == up) { keys[t] = c; keys[ixj] = a; }
                }
            }
            __syncthreads();
        }
    }

    // F = number of fg entries (boundary in sorted order; exactly one writer)
    if (tid == 0) sF = NP;
    __syncthreads();
    for (int i = tid; i < NP; i += P2_THREADS) {
        bool is_bg   = (keys[i] >> 62) & 1ull;
        bool prev_fg = (i == 0) ? true : !((keys[i - 1] >> 62) & 1ull);
        if (is_bg && prev_fg) sF = i;
    }
    __syncthreads();
    const int F = sF;

    // one sampled slot per thread
    const int s = tid;                         // [0, NS)
    int pos; bool valid;
    if (s < NUM_FG) {
        pos   = s;
        valid = (s < F);                       // fg slot valid iff a real fg entry
    } else {
        int j = s - NUM_FG;
        int q = F + j;
        valid = (q < NP);                      // bg slot valid iff a real bg entry
        pos   = valid ? q : (NP - 1);
    }
    const int idx = (int)(keys[pos] & 0xFFFFull);

    const float4 pbox = prop[b * NP + idx];
    const int    m    = matched_idxs[b * NP + idx];
    const float4 gbox = gt[b * NG + m];
    const float  mv   = matched_vals[b * NP + idx];
    const int    cls  = (mv >= 0.5f) ? gt_classes[b * NG + m] : NUM_CLASSES;

    const float msk = valid ? 1.0f : 0.0f;
    float4 o0 = make_float4(pbox.x * msk, pbox.y * msk, pbox.z * msk, pbox.w * msk);
    float4 o1 = make_float4(gbox.x * msk, gbox.y * msk, gbox.z * msk, gbox.w * msk);
    float4* op = (float4*)(out + (size_t)(b * NS + s) * 8);
    op[0] = o0;
    op[1] = o1;
    out[(size_t)NB * NS * 8 + b * NS + s] = valid ? (float)cls : -1.0f;
}

// ---------------------------------------------------------------------------
extern "C" void kernel_launch(void* const* d_in, const int* in_sizes, int n_in,
                              void* d_out, int out_size, void* d_ws, size_t ws_size,
                              hipStream_t stream)
{
    const float4* prop       = (const float4*)d_in[0];  // [16,4096,4] f32
    const float4* gt         = (const float4*)d_in[1];  // [16,512,4]  f32
    const int*    gt_classes = (const int*)  d_in[2];   // [16,512]    i32
    const float*  prio       = (const float*)d_in[3];   // [16,4096]   f32

    float* matched_vals = (float*)d_ws;                              // 256 KB
    int*   matched_idxs = (int*)((char*)d_ws + (size_t)NB * NP * 4); // 256 KB

    dim3 g1(NP / P1_THREADS, NB);
    roi_match_kernel<<<g1, P1_THREADS, 0, stream>>>(prop, gt, matched_vals, matched_idxs);

    roi_sample_kernel<<<NB, P2_THREADS, 0, stream>>>(prop, gt, gt_classes, prio,
                                                     matched_vals, matched_idxs,
                                                     (float*)d_out);
}